// overlap_singlet_fission_rate_24017457119682
// MI455X (gfx1250) — compile-verified
//
#include <hip/hip_runtime.h>

typedef float v2f __attribute__((ext_vector_type(2)));
typedef float v8f __attribute__((ext_vector_type(8)));

#define NAO 256

// One wave (32 lanes) per molecule/batch.
// Computes R = [hA;lA] (2x256) * S (256x256) * [hB,lB] (256x2) via
// V_WMMA_F32_16X16X4_F32, then t2 = (S10*S11 - S01*S00)^2.
__global__ __launch_bounds__(256) void sf_rate_wmma_kernel(
    const float* __restrict__ hA, const float* __restrict__ lA,
    const float* __restrict__ hB, const float* __restrict__ lB,
    const float* __restrict__ S,  float* __restrict__ out, int nmol)
{
    const int wavesPerBlock = blockDim.x >> 5;
    const int b    = blockIdx.x * wavesPerBlock + (threadIdx.x >> 5);
    if (b >= nmol) return;                    // lane-uniform: no divergence
    const int lane = threadIdx.x & 31;
    const int r    = lane & 15;               // N position / A row
    const int half = lane >> 4;               // K half for 16x4 A / 4x16 B

    const float* __restrict__ Sb  = S  + (size_t)b * NAO * NAO;
    const float* __restrict__ hBb = hB + (size_t)b * NAO;
    const float* __restrict__ lBb = lB + (size_t)b * NAO;
    // Row 0 of A = hA, row 1 = lA, rows 2..15 = 0 (masked multiply, no branch).
    const float* __restrict__ srcA = (r == 1) ? (lA + (size_t)b * NAO)
                                              : (hA + (size_t)b * NAO);
    const float amask = (r < 2) ? 1.0f : 0.0f;

    float s00 = 0.f, s01 = 0.f, s10 = 0.f, s11 = 0.f;

    for (int jt = 0; jt < 4; ++jt) {          // 4 passes x 64 columns each
        const int n0 = jt * 64;
        v8f acc0 = {}, acc1 = {}, acc2 = {}, acc3 = {};

        #pragma unroll 2
        for (int k = 0; k < NAO; k += 4) {
            // A tile 16x4: lane holds A[r][k+2*half], A[r][k+2*half+1]
            v2f a;
            a[0] = amask * srcA[k + 2 * half];
            a[1] = amask * srcA[k + 2 * half + 1];

            // B tiles 4x16: lanes 0-15 rows k..k+1, lanes 16-31 rows k+2..k+3
            const float* row0 = Sb + (size_t)(k + 2 * half)     * NAO;
            const float* row1 = Sb + (size_t)(k + 2 * half + 1) * NAO;
            v2f b0; b0[0] = row0[n0      + r]; b0[1] = row1[n0      + r];
            v2f b1; b1[0] = row0[n0 + 16 + r]; b1[1] = row1[n0 + 16 + r];
            v2f b2; b2[0] = row0[n0 + 32 + r]; b2[1] = row1[n0 + 32 + r];
            v2f b3; b3[0] = row0[n0 + 48 + r]; b3[1] = row1[n0 + 48 + r];

            acc0 = __builtin_amdgcn_wmma_f32_16x16x4_f32(false, a, false, b0,
                                                         (short)0, acc0, false, false);
            acc1 = __builtin_amdgcn_wmma_f32_16x16x4_f32(false, a, false, b1,
                                                         (short)0, acc1, false, false);
            acc2 = __builtin_amdgcn_wmma_f32_16x16x4_f32(false, a, false, b2,
                                                         (short)0, acc2, false, false);
            acc3 = __builtin_amdgcn_wmma_f32_16x16x4_f32(false, a, false, b3,
                                                         (short)0, acc3, false, false);
        }

        // Harvest: acc[0] = hS row (M=0) on lanes 0-15, acc[1] = lS row (M=1).
        // Lanes 16-31 hold rows M=8/9 which are exactly zero -> contribute 0.
        #pragma unroll
        for (int jj = 0; jj < 4; ++jj) {
            v8f acc = (jj == 0) ? acc0 : (jj == 1) ? acc1 : (jj == 2) ? acc2 : acc3;
            const int n = n0 + jj * 16 + r;
            const float hb = hBb[n];
            const float lb = lBb[n];
            const float hs = acc[0];
            const float ls = acc[1];
            s00 += hs * hb;
            s01 += hs * lb;
            s10 += ls * hb;
            s11 += ls * lb;
        }
    }

    // Wave-wide sum (lanes 16-31 carry zeros; reducing all 32 lanes is fine).
    #pragma unroll
    for (int off = 16; off > 0; off >>= 1) {
        s00 += __shfl_xor(s00, off, 32);
        s01 += __shfl_xor(s01, off, 32);
        s10 += __shfl_xor(s10, off, 32);
        s11 += __shfl_xor(s11, off, 32);
    }

    if (lane == 0) {
        const float d = s10 * s11 - s01 * s00;
        out[b] = d * d;
    }
}

extern "C" void kernel_launch(void* const* d_in, const int* in_sizes, int n_in,
                              void* d_out, int out_size, void* d_ws, size_t ws_size,
                              hipStream_t stream) {
    const float* hA = (const float*)d_in[0];
    const float* lA = (const float*)d_in[1];
    const float* hB = (const float*)d_in[2];
    const float* lB = (const float*)d_in[3];
    const float* S  = (const float*)d_in[4];
    float* out = (float*)d_out;

    const int nmol = in_sizes[0] / NAO;       // 4096
    const int wavesPerBlock = 8;              // 256 threads = 8 wave32
    dim3 block(256);
    dim3 grid((nmol + wavesPerBlock - 1) / wavesPerBlock);
    sf_rate_wmma_kernel<<<grid, block, 0, stream>>>(hA, lA, hB, lB, S, out, nmol);
}